// SptialTopk_62328565399656
// MI455X (gfx1250) — compile-verified
//
#include <hip/hip_runtime.h>

typedef __attribute__((ext_vector_type(2))) float v2f;
typedef __attribute__((ext_vector_type(8))) float v8f;

#define BB   256
#define NN   4096
#define DIM  64
#define TOPKK 1024
#define NHH  8
#define WINW 49

// map float -> unsigned so that larger float => larger key (no NaNs in input)
__device__ __forceinline__ unsigned okey(float f) {
  unsigned u = __float_as_uint(f);
  return u ^ (((int)u < 0) ? 0xFFFFFFFFu : 0x80000000u);
}

__global__ __launch_bounds__(256)
void spatial_topk_gather_kernel(const float* __restrict__ q,
                                const float* __restrict__ qq,
                                const float* __restrict__ bias,
                                float* __restrict__ out) {
  __shared__ float    xs[NN];        // 16 KB: row means for this batch
  __shared__ unsigned hist[256];     // radix histogram
  __shared__ unsigned gts[256];      // per-thread ">threshold" counts -> prefix
  __shared__ unsigned eqs[256];      // per-thread "==threshold" counts -> prefix
  __shared__ int      p_sh[TOPKK];   // 4 KB: selected indices, ascending
  __shared__ unsigned s_pref, s_krem;

  const int b    = blockIdx.x;
  const int t    = threadIdx.x;
  const int lane = t & 31;
  const int wave = t >> 5;

  const float* qb  = q  + (size_t)b * NN * DIM;
  const float* qqb = qq + (size_t)b * NN * DIM;

  // ---------------- Phase 1: row means via v_wmma_f32_16x16x4_f32 ----------
  // D(16x16) += A(16x4) * B(4x16), B == 1/64 everywhere -> every column of D
  // is the running mean of the corresponding A row.  16 chunks cover K=64.
  {
    const int m    = lane & 15;   // A-matrix M for this lane
    const int kgrp = lane >> 4;   // lanes 0-15: K={0,1}; lanes 16-31: K={2,3}
    v2f bfrag; bfrag[0] = 1.0f / 64.0f; bfrag[1] = 1.0f / 64.0f;
    for (int g = 0; g < 32; ++g) {                 // 8 waves * 32 groups * 16 rows = 4096
      const int row0 = wave * 512 + g * 16;
      const float* rp = qb + (size_t)(row0 + m) * DIM + 2 * kgrp;
      v8f acc = {};
      #pragma unroll
      for (int c = 0; c < 16; ++c) {
        v2f a;
        a[0] = rp[4 * c + 0];
        a[1] = rp[4 * c + 1];
        acc = __builtin_amdgcn_wmma_f32_16x16x4_f32(false, a, false, bfrag,
                                                    (short)0, acc, false, false);
      }
      // D layout: lanes 0-15 hold M=0..7 in v0..v7; lanes 16-31 hold M=8..15.
      if (lane == 0) {
        #pragma unroll
        for (int r = 0; r < 8; ++r) xs[row0 + r] = acc[r];
      } else if (lane == 16) {
        #pragma unroll
        for (int r = 0; r < 8; ++r) xs[row0 + 8 + r] = acc[r];
      }
    }
  }
  if (t == 0) { s_pref = 0u; s_krem = TOPKK; }
  __syncthreads();

  // ---------------- Phase 2: radix-select the 1024th largest ---------------
  for (int pass = 0; pass < 4; ++pass) {
    const int shift = 24 - 8 * pass;
    hist[t] = 0u;
    __syncthreads();
    const unsigned pref = s_pref;
    const unsigned mask = (pass == 0) ? 0u : (0xFFFFFFFFu << (shift + 8));
    #pragma unroll 4
    for (int j = 0; j < 16; ++j) {
      const unsigned u = okey(xs[t * 16 + j]);
      if ((u & mask) == pref) atomicAdd(&hist[(u >> shift) & 255u], 1u);
    }
    __syncthreads();
    if (t == 0) {
      const unsigned krem = s_krem;
      unsigned cum = 0u; int d = 255;
      for (; d > 0; --d) {
        const unsigned h = hist[d];
        if (cum + h >= krem) break;
        cum += h;
      }
      s_pref = pref | ((unsigned)d << shift);
      s_krem = krem - cum;
    }
    __syncthreads();
  }
  const unsigned Tkey  = s_pref;   // key of the 1024th-largest mean
  const unsigned Tneed = s_krem;   // how many ties (lowest index first) to keep

  // ---------------- stable index-ordered compaction ------------------------
  {
    unsigned gt = 0u, eq = 0u;
    #pragma unroll 4
    for (int j = 0; j < 16; ++j) {
      const unsigned u = okey(xs[t * 16 + j]);
      gt += (u > Tkey);
      eq += (u == Tkey);
    }
    gts[t] = gt; eqs[t] = eq;
    __syncthreads();
    if (t == 0) {                  // exclusive scans (tiny; once per block)
      unsigned ga = 0u, ea = 0u;
      for (int i = 0; i < 256; ++i) {
        const unsigned g = gts[i], e = eqs[i];
        gts[i] = ga; eqs[i] = ea;
        ga += g; ea += e;
      }
    }
    __syncthreads();
    unsigned gb = gts[t], eb = eqs[t];
    for (int j = 0; j < 16; ++j) {
      const int i = t * 16 + j;
      const unsigned u = okey(xs[i]);
      if (u > Tkey) {
        const unsigned pos = gb + (eb < Tneed ? eb : Tneed);
        p_sh[pos] = i;
        ++gb;
      } else if (u == Tkey) {
        if (eb < Tneed) p_sh[gb + eb] = i;
        ++eb;
      }
    }
  }
  __syncthreads();

  // ---------------- Phase 3a: gather qal / val (256B rows, float4) ---------
  const size_t QAL_TOT = (size_t)BB * TOPKK * DIM;   // 16,777,216
  {
    const int sub = t & 15;        // float4 slot within a 64-float row
    const int rk  = t >> 4;        // 16 rows gathered per iteration
    float* out_qal = out;
    float* out_val = out + QAL_TOT;
    for (int kk = rk; kk < TOPKK; kk += 16) {
      const int src = p_sh[kk];
      const float4 vqq = *(const float4*)(qqb + (size_t)src * DIM + sub * 4);
      const float4 vq  = *(const float4*)(qb  + (size_t)src * DIM + sub * 4);
      const size_t o = ((size_t)b * TOPKK + kk) * DIM + sub * 4;
      *(float4*)(out_qal + o) = vqq;
      *(float4*)(out_val + o) = vq;
    }
  }

  // ---------------- Phase 3b: gather yal from bias --------------------------
  // yal[bh][h][w][k] = bias[h][w][p[b][k]],  b = bh*8 + h
  {
    const int bh = b >> 3;
    const int h  = b & 7;
    float* out_yal = out + 2 * QAL_TOT;
    const float*  brow0 = bias + (size_t)h * WINW * NN;
    const size_t  obase = (size_t)bh * (NHH * WINW * TOPKK) + (size_t)h * (WINW * TOPKK);
    for (int w = 0; w < WINW; ++w) {
      const float* br = brow0 + (size_t)w * NN;
      if (w + 1 < WINW) __builtin_prefetch(br + NN, 0, 0);   // warm next 16KB bias row
      const size_t ob = obase + (size_t)w * TOPKK;
      #pragma unroll 4
      for (int kk = t; kk < TOPKK; kk += 256)
        out_yal[ob + kk] = br[p_sh[kk]];
    }
  }
}

extern "C" void kernel_launch(void* const* d_in, const int* in_sizes, int n_in,
                              void* d_out, int out_size, void* d_ws, size_t ws_size,
                              hipStream_t stream) {
  const float* q    = (const float*)d_in[0];   // (256, 4096, 64)
  const float* qq   = (const float*)d_in[1];   // (256, 4096, 64)
  const float* bias = (const float*)d_in[2];   // (8, 49, 4096)
  float* out = (float*)d_out;                  // qal | val | yal, flat

  spatial_topk_gather_kernel<<<dim3(BB), dim3(256), 0, stream>>>(q, qq, bias, out);
}